// SubPixelRefinement_63050119905589
// MI455X (gfx1250) — compile-verified
//
#include <hip/hip_runtime.h>
#include <stdint.h>

// Problem constants from the reference: heatmaps (256, 17, 128, 96) f32, RADIUS=2, BETA=1.
#define HM_H 128
#define HM_W 96
#define HM_HW (HM_H * HM_W)           // 12288 floats = 48 KB per map
#define BLOCK 256                     // 8 wave32 per workgroup
#define NWAVES (BLOCK / 32)
#define VECS_PER_THREAD (HM_HW / (BLOCK * 4))  // 12 float4 per thread
#define RADIUS 2

// ---------------- kernel: one workgroup per (b,k) heatmap --------------------------------
__global__ __launch_bounds__(BLOCK)
void subpixel_refine_kernel(const float* __restrict__ hm,
                            const float* __restrict__ alpha_p,
                            float* __restrict__ out,
                            int nmaps) {
    __shared__ __align__(16) float s_hm[HM_HW];
    __shared__ float s_max[NWAVES];
    __shared__ float s_sum[NWAVES][3];   // sumE, sumEx, sumEy per wave

    const int bk = blockIdx.x;
    const int t  = threadIdx.x;
    const float* __restrict__ src = hm + (size_t)bk * HM_HW;

    // ---- Phase 0: async DMA global -> LDS (CDNA5 ASYNCcnt path), 16 B per lane per issue.
    // AS3 byte address of an LDS object == low 32 bits of its generic pointer on amdgcn.
    {
        const uint64_t gbase = (uint64_t)(uintptr_t)src;
        const uint32_t lbase = (uint32_t)(uintptr_t)(&s_hm[0]);
#pragma unroll
        for (int i = 0; i < VECS_PER_THREAD; ++i) {
            uint32_t off   = (uint32_t)((i * BLOCK + t) * 16);
            uint32_t laddr = lbase + off;
            uint64_t gaddr = gbase + off;
            asm volatile("global_load_async_to_lds_b128 %0, %1, off"
                         :: "v"(laddr), "v"(gaddr)
                         : "memory");
        }
    }
#if __has_builtin(__builtin_amdgcn_s_wait_asynccnt)
    __builtin_amdgcn_s_wait_asynccnt(0);
#else
    asm volatile("s_wait_asynccnt 0x0" ::: "memory");
#endif
    // No barrier needed yet: passes 1/2 only touch this thread's own staged data, and every
    // wave has drained its own async queue before the first reduction barrier below.

    const float4* __restrict__ s4 = reinterpret_cast<const float4*>(s_hm);
    const int wave = t >> 5;
    const int lane = t & 31;

    // ---- Pass 1: block-wide max (branchless; also the output score).
    float m = -3.402823466e38f;
#pragma unroll
    for (int i = 0; i < VECS_PER_THREAD; ++i) {
        float4 v = s4[i * BLOCK + t];
        m = fmaxf(m, fmaxf(fmaxf(v.x, v.y), fmaxf(v.z, v.w)));
    }
#pragma unroll
    for (int off = 16; off >= 1; off >>= 1)
        m = fmaxf(m, __shfl_xor(m, off));
    if (lane == 0) s_max[wave] = m;
    __syncthreads();

    float M = s_max[0];
#pragma unroll
    for (int wv = 1; wv < NWAVES; ++wv)
        M = fmaxf(M, s_max[wv]);

    // ---- Pass 2: e = exp(v - M); accumulate sumE, sumE*x, sumE*y (branchless FMA form).
    float sumE = 0.0f, sx = 0.0f, sy = 0.0f;
#pragma unroll
    for (int i = 0; i < VECS_PER_THREAD; ++i) {
        int j = i * BLOCK + t;
        float4 v = s4[j];
        int flat = j * 4;                  // 96 % 4 == 0 -> float4 never crosses a row
        int h  = flat / HM_W;
        int w0 = flat - h * HM_W;
        float e0 = __expf(v.x - M);
        float e1 = __expf(v.y - M);
        float e2 = __expf(v.z - M);
        float e3 = __expf(v.w - M);
        float rowE = (e0 + e1) + (e2 + e3);
        float frac = fmaf(2.0f, e2, e1);   // e1 + 2*e2
        frac = fmaf(3.0f, e3, frac);       // + 3*e3
        sumE += rowE;
        sx = fmaf((float)w0, rowE, sx) + frac;   // sum e*(w0 + c)
        sy = fmaf((float)h,  rowE, sy);          // sum e*h
    }

    // wave32 butterfly reduction of the three sums
#pragma unroll
    for (int off = 16; off >= 1; off >>= 1) {
        sumE += __shfl_xor(sumE, off);
        sx   += __shfl_xor(sx,   off);
        sy   += __shfl_xor(sy,   off);
    }
    if (lane == 0) {
        s_sum[wave][0] = sumE;
        s_sum[wave][1] = sx;
        s_sum[wave][2] = sy;
    }
    __syncthreads();

    // ---- Phase 3: final merge + 5x5 masked-softmax refinement (tiny serial tail).
    if (t == 0) {
        float S = 0.0f, SX = 0.0f, SY = 0.0f;
#pragma unroll
        for (int wv = 0; wv < NWAVES; ++wv) {
            S  += s_sum[wv][0];
            SX += s_sum[wv][1];
            SY += s_sum[wv][2];
        }
        const float gx    = SX / S;        // soft-argmax x
        const float gy    = SY / S;        // soft-argmax y
        const float score = M;             // max over map

        // round-half-even matches jnp.round; clip to bounds
        int px = (int)rintf(gx); px = px < 0 ? 0 : (px > HM_W - 1 ? HM_W - 1 : px);
        int py = (int)rintf(gy); py = py < 0 ? 0 : (py > HM_H - 1 ? HM_H - 1 : py);

        // masked softmax over the (2r+1)^2 window; out-of-bounds entries get weight 0
        float lm = -3.402823466e38f;
#pragma unroll
        for (int dy = -RADIUS; dy <= RADIUS; ++dy) {
#pragma unroll
            for (int dx = -RADIUS; dx <= RADIUS; ++dx) {
                int xs = px + dx, ys = py + dy;
                if (xs >= 0 && xs < HM_W && ys >= 0 && ys < HM_H)
                    lm = fmaxf(lm, s_hm[ys * HM_W + xs]);
            }
        }
        float ls = 0.0f, lsx = 0.0f, lsy = 0.0f;
#pragma unroll
        for (int dy = -RADIUS; dy <= RADIUS; ++dy) {
#pragma unroll
            for (int dx = -RADIUS; dx <= RADIUS; ++dx) {
                int xs = px + dx, ys = py + dy;
                if (xs >= 0 && xs < HM_W && ys >= 0 && ys < HM_H) {
                    float e = __expf(s_hm[ys * HM_W + xs] - lm);
                    ls  += e;
                    lsx = fmaf(e, (float)xs, lsx);
                    lsy = fmaf(e, (float)ys, lsy);
                }
            }
        }
        const float lx = lsx / ls;
        const float ly = lsy / ls;

        const float a = 1.0f / (1.0f + __expf(-alpha_p[0]));   // sigmoid(alpha)
        // outputs: refined (B,K,2) flat, then scores (B,K) flat
        out[bk * 2 + 0]     = a * gx + (1.0f - a) * lx;
        out[bk * 2 + 1]     = a * gy + (1.0f - a) * ly;
        out[nmaps * 2 + bk] = score;
    }
}

extern "C" void kernel_launch(void* const* d_in, const int* in_sizes, int n_in,
                              void* d_out, int out_size, void* d_ws, size_t ws_size,
                              hipStream_t stream) {
    const float* hm    = (const float*)d_in[0];   // (256,17,128,96) f32
    const float* alpha = (const float*)d_in[1];   // scalar f32
    float* out = (float*)d_out;                   // refined (B,K,2) ++ scores (B,K)

    const int nmaps = in_sizes[0] / HM_HW;        // 256*17 = 4352
    subpixel_refine_kernel<<<dim3(nmaps), dim3(BLOCK), 0, stream>>>(hm, alpha, out, nmaps);
}